// HeapLinear_71571335020654
// MI455X (gfx1250) — compile-verified
//
#include <hip/hip_runtime.h>

typedef __attribute__((ext_vector_type(2))) float v2f;
typedef __attribute__((ext_vector_type(8))) float v8f;

#define NODES 16384
#define IN_F  64
#define OUT_F 64
#define BATCH 32
#define DEPTH 15

// One wave (32 lanes) per (node j, batch-half). Computes out[b0:b0+16, j, 0:64]
// = sum_{k=0..bit_length(j)} x[b0:b0+16, j>>k, :] @ W[k]^T  + csum_bias[bit_length(j)]
// using v_wmma_f32_16x16x4_f32, accumulating K=64 in 16 steps, 4 N-tiles.
__global__ __launch_bounds__(256) void heaplinear_wmma_kernel(
    const float* __restrict__ x,     // [BATCH, NODES, IN_F]
    const float* __restrict__ W,     // [DEPTH, OUT_F, IN_F]
    const float* __restrict__ bias,  // [DEPTH, OUT_F]
    float* __restrict__ out)         // [BATCH, NODES, OUT_F]
{
    const int lane = threadIdx.x & 31;
    const int wid  = threadIdx.x >> 5;
    const int task = blockIdx.x * 8 + wid;   // 2*NODES wave-tasks
    const int j    = task >> 1;              // heap node
    const int b0   = (task & 1) << 4;        // batch half: 0 or 16

    const int M     = lane & 15;   // A-row (batch) / B-row (out feat) / D N-index
    const int khalf = lane >> 4;   // 0 -> K={0,1}, 1 -> K={2,3} within each 4-slice

    // level = bit_length(j) - 1 (0 for j == 0); wave-uniform (j is wave-uniform)
    const int level = (j == 0) ? 0 : (32 - __builtin_clz((unsigned)j) - 1 + 1) - 1 + 1 - 1;
    // (bit_length(j) for j>0 is 32-clz; masked terms are k=0..bit_length)
    const int nterms = (j == 0) ? 1 : (32 - __builtin_clz((unsigned)j)) + 1;

    // Fold cumulative bias into accumulator init: D = A*B + C.
    v8f acc[4];
#pragma unroll
    for (int t = 0; t < 4; ++t) {
        float cs = 0.0f;
        const int o = t * 16 + M;                 // N index (out feature)
        for (int k = 0; k < nterms; ++k)          // wave-uniform trip count
            cs += bias[k * OUT_F + o];
        acc[t] = (v8f){cs, cs, cs, cs, cs, cs, cs, cs};
    }

    int node = j;
    for (int k = 0; k < nterms; ++k) {
        // A fragments: x row for batch (b0+M), ancestor node, this lane's K-half.
        const float* xr = x + ((size_t)(b0 + M) * NODES + node) * IN_F + 2 * khalf;
        v2f afrag[16];
#pragma unroll
        for (int s = 0; s < 16; ++s)
            afrag[s] = *(const v2f*)(xr + 4 * s);  // 8B aligned b64 load

        const float* wk = W + (size_t)k * OUT_F * IN_F;
#pragma unroll
        for (int t = 0; t < 4; ++t) {
            // B = W[k]^T tile: element (K=i, N=o) = W[k][o][i]
            const float* wr = wk + (t * 16 + M) * IN_F + 2 * khalf;
#pragma unroll
            for (int s = 0; s < 16; ++s) {
                v2f bfrag = *(const v2f*)(wr + 4 * s);
                acc[t] = __builtin_amdgcn_wmma_f32_16x16x4_f32(
                    false, afrag[s], false, bfrag,
                    (short)0, acc[t], false, false);
            }
        }
        node >>= 1;
    }

    // D layout 16x16 f32: VGPR r, lanes 0-15 -> (M=r, N=lane); lanes 16-31 -> (M=8+r, N=lane-16)
#pragma unroll
    for (int t = 0; t < 4; ++t) {
        const int o = t * 16 + M;
#pragma unroll
        for (int r = 0; r < 8; ++r) {
            const int brow = b0 + r + 8 * khalf;
            out[((size_t)brow * NODES + j) * OUT_F + o] = acc[t][r];
        }
    }
    (void)level;
}

extern "C" void kernel_launch(void* const* d_in, const int* in_sizes, int n_in,
                              void* d_out, int out_size, void* d_ws, size_t ws_size,
                              hipStream_t stream) {
    const float* x = (const float*)d_in[0];   // [32, 16384, 64] f32
    const float* W = (const float*)d_in[1];   // [15, 64, 64]    f32
    const float* b = (const float*)d_in[2];   // [15, 64]        f32
    float* out = (float*)d_out;               // [32, 16384, 64] f32

    const int tasks = 2 * NODES;              // 32768 wave-tasks
    dim3 grid(tasks / 8);                     // 8 waves (256 threads) per block
    heaplinear_wmma_kernel<<<grid, 256, 0, stream>>>(x, W, b, out);
}